// GraphSAGE_87144886436073
// MI455X (gfx1250) — compile-verified
//
#include <hip/hip_runtime.h>

// ---------------------------------------------------------------------------
// GraphSAGE (3 layers, D=64) for MI455X (gfx1250, wave32).
// Scatter phase: hardware f32 global atomics (L2-resident feature table).
// Dense phase: V_WMMA_F32_16X16X4_F32, one wave = 16-node tile.
// ---------------------------------------------------------------------------

#define N_NODES  100000
#define N_EDGES  1600000
#define D        64
#define N_GRAPHS 64

typedef __attribute__((ext_vector_type(2))) float v2f;
typedef __attribute__((ext_vector_type(8))) float v8f;

// ---------------------------------------------------------------- zero fill
__global__ void k_zero_f32(float* __restrict__ p, int n) {
    int i = blockIdx.x * blockDim.x + threadIdx.x;
    int stride = gridDim.x * blockDim.x;
    for (; i < n; i += stride) p[i] = 0.0f;
}

// ---------------------------------------------------------------- in-degree
__global__ void k_count_deg(const int* __restrict__ dst, float* __restrict__ cnt) {
    int e = blockIdx.x * blockDim.x + threadIdx.x;
    if (e < N_EDGES) unsafeAtomicAdd(&cnt[dst[e]], 1.0f);
}

__global__ void k_invert_deg(float* __restrict__ c) {
    int i = blockIdx.x * blockDim.x + threadIdx.x;
    if (i < N_NODES) {
        float v = c[i];
        c[i] = (v > 0.0f) ? (1.0f / v) : 0.0f;
    }
}

// ------------------------------------------------------- edge scatter (sum)
// 32 threads per edge; each thread moves 2 consecutive floats of the 64-wide
// row: coalesced 256B gather per edge + 2 hardware f32 atomics per lane.
__global__ void k_scatter_edges(const float* __restrict__ h,
                                const int* __restrict__ src,
                                const int* __restrict__ dst,
                                float* __restrict__ agg) {
    long long tid = (long long)blockIdx.x * blockDim.x + threadIdx.x;
    int e = (int)(tid >> 5);
    if (e >= N_EDGES) return;
    int j = ((int)tid & 31) * 2;
    int s = src[e];
    int d = dst[e];
    const float2 v = *(const float2*)(h + (size_t)s * D + j);
    float* out = agg + (size_t)d * D + j;
    unsafeAtomicAdd(out,     v.x);
    unsafeAtomicAdd(out + 1, v.y);
}

// --------------------------------------------- dense layer: WMMA f32 16x16x4
// hout[n, :] = (agg[n,:] * dinv[n]) @ Wl^T + bl + hin[n,:] @ Wr^T  (opt. ReLU)
//
// One wave per 16-node tile. A-fragment (16x4 f32, 2 VGPR/lane):
//   lane L: row = L%16, K-offset = 2*(L/16); a = {A[row][k0+off], A[row][k0+off+1]}
// B-fragment (4x16): B[k][o] = W[o][k]  ->  same float2 addressing into W.
// C/D layout: VGPR r -> M = r + 8*(L/16), N = L%16.
__global__ void __launch_bounds__(256)
k_sage_apply(const float* __restrict__ agg,
             const float* __restrict__ dinv,
             const float* __restrict__ hin,
             const float* __restrict__ Wl,
             const float* __restrict__ bl,
             const float* __restrict__ Wr,
             float* __restrict__ hout,
             int do_relu) {
    const int waveInBlock = threadIdx.x >> 5;
    const int waveId = blockIdx.x * (blockDim.x >> 5) + waveInBlock;
    const int nTiles = N_NODES / 16;           // 6250, exact
    if (waveId >= nTiles) return;              // uniform per wave: EXEC stays all-1s

    const int lane = threadIdx.x & 31;
    const int row  = lane & 15;                // M index (A) / N index (B,C,D)
    const int half = lane >> 4;                // lane-half selects K-pair / M+8
    const int base = waveId * 16;              // first node of tile

    // Preload A fragments for all 16 K-steps (K = 64 = 16 * 4)
    const float dv = dinv[base + row];
    const float* aggRow = agg + (size_t)(base + row) * D + 2 * half;
    const float* hRow   = hin + (size_t)(base + row) * D + 2 * half;
    v2f aAgg[16], aH[16];
#pragma unroll
    for (int k = 0; k < 16; ++k) {
        v2f t  = *(const v2f*)(aggRow + 4 * k);
        aAgg[k] = t * dv;                       // fold deg_inv into A
        aH[k]   = *(const v2f*)(hRow + 4 * k);
    }

    // 4 output column tiles of 16
#pragma unroll
    for (int nt = 0; nt < 4; ++nt) {
        const int n0 = nt * 16;
        const float bb = bl[n0 + row];
        v8f acc = {bb, bb, bb, bb, bb, bb, bb, bb};   // bias depends only on N

        const float* wlP = Wl + (size_t)(n0 + row) * D + 2 * half;
        const float* wrP = Wr + (size_t)(n0 + row) * D + 2 * half;
#pragma unroll
        for (int k = 0; k < 16; ++k) {
            v2f bL = *(const v2f*)(wlP + 4 * k);
            v2f bR = *(const v2f*)(wrP + 4 * k);
            acc = __builtin_amdgcn_wmma_f32_16x16x4_f32(
                false, aAgg[k], false, bL, (short)0, acc, false, false);
            acc = __builtin_amdgcn_wmma_f32_16x16x4_f32(
                false, aH[k],   false, bR, (short)0, acc, false, false);
        }

#pragma unroll
        for (int r = 0; r < 8; ++r) {
            float v = acc[r];
            if (do_relu) v = fmaxf(v, 0.0f);
            const int node = base + r + 8 * half;
            hout[(size_t)node * D + n0 + row] = v;
        }
    }
}

// ---------------------------------------------------------- graph pooling
__global__ void k_pool(const float* __restrict__ h,
                       const int* __restrict__ batch,
                       float* __restrict__ out) {
    int tid = blockIdx.x * blockDim.x + threadIdx.x;
    if (tid >= N_NODES * D) return;
    int node = tid >> 6;
    int col  = tid & 63;
    unsafeAtomicAdd(&out[batch[node] * D + col], h[tid]);
}

// ---------------------------------------------------------------------------
extern "C" void kernel_launch(void* const* d_in, const int* in_sizes, int n_in,
                              void* d_out, int out_size, void* d_ws, size_t ws_size,
                              hipStream_t stream) {
    const float* x     = (const float*)d_in[0];
    const int*   eidx  = (const int*)d_in[1];
    const int*   batch = (const int*)d_in[2];
    const float* Wl[3] = {(const float*)d_in[3], (const float*)d_in[6], (const float*)d_in[9]};
    const float* bl[3] = {(const float*)d_in[4], (const float*)d_in[7], (const float*)d_in[10]};
    const float* Wr[3] = {(const float*)d_in[5], (const float*)d_in[8], (const float*)d_in[11]};
    const int* src = eidx;
    const int* dst = eidx + N_EDGES;
    float* out = (float*)d_out;

    // Workspace layout (floats), 256-element aligned chunks
    float* ws = (float*)d_ws;
    const size_t dinvOff = 0;
    const size_t dinvLen = ((N_NODES + 255) / 256) * 256;        // 100096
    const size_t featLen = (size_t)N_NODES * D;                  // 6.4M floats
    float* dinv = ws + dinvOff;
    float* agg  = ws + dinvLen;
    float* h1   = agg + featLen;
    float* h2   = h1 + featLen;

    const int TPB = 256;

    // Degree + inverse
    k_zero_f32<<<1024, TPB, 0, stream>>>(dinv, (int)dinvLen);
    k_count_deg<<<(N_EDGES + TPB - 1) / TPB, TPB, 0, stream>>>(dst, dinv);
    k_invert_deg<<<(N_NODES + TPB - 1) / TPB, TPB, 0, stream>>>(dinv);

    const int scatterBlocks = (int)(((long long)N_EDGES * 32 + TPB - 1) / TPB);
    const int applyBlocks   = (N_NODES / 16 + 7) / 8;            // 8 waves/block

    const float* hin = x;
    float* hbufs[3] = {h1, h2, h1};
    for (int L = 0; L < 3; ++L) {
        float* hout = hbufs[L];
        k_zero_f32<<<2048, TPB, 0, stream>>>(agg, (int)featLen);
        k_scatter_edges<<<scatterBlocks, TPB, 0, stream>>>(hin, src, dst, agg);
        k_sage_apply<<<applyBlocks, TPB, 0, stream>>>(agg, dinv, hin,
                                                      Wl[L], bl[L], Wr[L],
                                                      hout, (L < 2) ? 1 : 0);
        hin = hout;
    }

    // Graph pooling
    k_zero_f32<<<16, TPB, 0, stream>>>(out, N_GRAPHS * D);
    k_pool<<<(N_NODES * D + TPB - 1) / TPB, TPB, 0, stream>>>(hbufs[2], batch, out);
}